// BiLingual_44341242364621
// MI455X (gfx1250) — compile-verified
//
#include <hip/hip_runtime.h>

typedef __attribute__((ext_vector_type(2))) float v2f;
typedef __attribute__((ext_vector_type(8))) float v8f;

#define SEQ 200
#define DIM 64

// Grid: (nrows, 2). blockIdx.y selects the table (scalar -> SGPR base for
// GVS-mode global_load saddr addressing). 4 wave32 per block, one 16-dim
// tile each. Each wave reduces (row, table, tile) with
// V_WMMA_F32_16X16X4_F32 using an all-ones A matrix:
//   D = ones(16x4) x B(4x16) + C  =>  D[0][n] = bag-sum of 4 tokens, dim n.
__global__ __launch_bounds__(128) void BiLingual_embedbag_wmma(
    const int* __restrict__ idx_pri,
    const int* __restrict__ idx_sec,
    const float* __restrict__ emb_pri,
    const float* __restrict__ emb_sec,
    float* __restrict__ out,
    int nrows)
{
    __shared__ int sidx[SEQ];

    const int row   = blockIdx.x;
    const int table = blockIdx.y;            // 0 = pri, 1 = sec (scalar)
    const int tid   = threadIdx.x;

    const int*   __restrict__ idx = (table == 0) ? idx_pri : idx_sec;
    const char*  __restrict__ emb =
        (const char*)((table == 0) ? emb_pri : emb_sec);

    // Stage this row's 200 indices into LDS once per block.
    for (int i = tid; i < SEQ; i += 128) sidx[i] = idx[row * SEQ + i];
    __syncthreads();

    const int wave     = tid >> 5;           // dim-tile 0..3
    const int lane     = tid & 31;
    const int laneIn   = lane & 15;          // N index within tile
    const int laneHalf = lane >> 4;          // K={0,1} vs K={2,3} half

    // 32-bit unsigned byte offset -> matches SADDR64 + zext(VOFF32) form.
    const unsigned colByte = (unsigned)(wave * 16 + laneIn) * 4u;

    const v2f a = {1.0f, 1.0f};              // ones A-matrix (16x4)
    v8f c0 = {};
    v8f c1 = {};

    // 200 tokens, 8 per iteration on 2 independent accumulators.
    #pragma unroll 5
    for (int k = 0; k < SEQ; k += 8) {
        // B layout 4x16 f32: VGPR0 = K0 (lanes<16) / K2 (lanes>=16),
        //                    VGPR1 = K1 / K3.   row stride = 64*4 = 256 B
        const unsigned o0 = ((unsigned)sidx[k + 0 + 2 * laneHalf] << 8) + colByte;
        const unsigned o1 = ((unsigned)sidx[k + 1 + 2 * laneHalf] << 8) + colByte;
        const unsigned o2 = ((unsigned)sidx[k + 4 + 2 * laneHalf] << 8) + colByte;
        const unsigned o3 = ((unsigned)sidx[k + 5 + 2 * laneHalf] << 8) + colByte;

        v2f b0, b1;
        b0.x = *(const float*)(emb + o0);
        b0.y = *(const float*)(emb + o1);
        b1.x = *(const float*)(emb + o2);
        b1.y = *(const float*)(emb + o3);

        c0 = __builtin_amdgcn_wmma_f32_16x16x4_f32(
                 false, a, false, b0, (short)0, c0, false, false);
        c1 = __builtin_amdgcn_wmma_f32_16x16x4_f32(
                 false, a, false, b1, (short)0, c1, false, false);
    }

    // C/D layout: VGPR0, lanes 0..15 hold D[M=0][N=lane] = full bag-sum.
    if (lane < 16) {
        out[(size_t)table * (size_t)nrows * DIM +
            (size_t)row * DIM + (wave * 16 + laneIn)] = c0[0] + c1[0];
    }
}

extern "C" void kernel_launch(void* const* d_in, const int* in_sizes, int n_in,
                              void* d_out, int out_size, void* d_ws, size_t ws_size,
                              hipStream_t stream) {
    const int*   idx_pri = (const int*)d_in[0];
    const int*   idx_sec = (const int*)d_in[1];
    const float* emb_pri = (const float*)d_in[2];
    const float* emb_sec = (const float*)d_in[3];
    float* out = (float*)d_out;

    const int nrows = in_sizes[0] / SEQ;     // 4096

    hipLaunchKernelGGL(BiLingual_embedbag_wmma,
                       dim3(nrows, 2), dim3(128), 0, stream,
                       idx_pri, idx_sec, emb_pri, emb_sec, out, nrows);
}